// FeaturePropagation_33569464385556
// MI455X (gfx1250) — compile-verified
//
#include <hip/hip_runtime.h>
#include <math.h>

// ---------------------------------------------------------------------------
// PointNet++ FeaturePropagation for MI455X (gfx1250, wave32, WMMA).
// Sizes fixed per reference: B=8, N=8192, M=1024, C=256, MLP=[256,256,128], K=3
// ---------------------------------------------------------------------------
#define B_   8
#define N_   8192
#define M_   1024
#define C_   256
#define P_   (B_ * N_)          // 65536 positions (B*N)
#define O2_  128                // last layer channels
#define BN_EPS 1e-5f

typedef __attribute__((ext_vector_type(2))) float v2f;
typedef __attribute__((ext_vector_type(4))) float v4f;
typedef __attribute__((ext_vector_type(8))) float v8f;

// ---------------------------------------------------------------------------
// Zero the per-layer stat accumulators (6 x 256 floats).
// ---------------------------------------------------------------------------
__global__ void fp_zero_stats(float* __restrict__ s) {
    int t = blockIdx.x * 256 + threadIdx.x;
    if (t < 1536) s[t] = 0.0f;
}

// ---------------------------------------------------------------------------
// Copy q (B*M*3 floats) to the front of d_out.
// ---------------------------------------------------------------------------
__global__ void fp_copy_q(const float* __restrict__ q, float* __restrict__ out) {
    int t = blockIdx.x * 256 + threadIdx.x;            // 6144 float4 quads
    ((float4*)out)[t] = ((const float4*)q)[t];
}

// ---------------------------------------------------------------------------
// 3-NN + inverse-distance weights. One thread per p-point; all q points of
// the batch staged in LDS (16 KB). Branchless sorted-insert keeps jax.top_k
// tie order (strict <).
// grid = (N/256, B), block = 256
// ---------------------------------------------------------------------------
__global__ __launch_bounds__(256) void fp_knn(
    const float* __restrict__ p, const float* __restrict__ q,
    int* __restrict__ idxO, float* __restrict__ wO)
{
    __shared__ float qx[M_], qy[M_], qz[M_], qq[M_];
    const int b = blockIdx.y;
    const float* qb = q + (size_t)b * M_ * 3;
    for (int m = threadIdx.x; m < M_; m += 256) {
        float x = qb[m * 3 + 0], y = qb[m * 3 + 1], z = qb[m * 3 + 2];
        qx[m] = x; qy[m] = y; qz[m] = z; qq[m] = x * x + y * y + z * z;
    }
    __syncthreads();

    const int n = blockIdx.x * 256 + threadIdx.x;
    const float* pr = p + ((size_t)b * N_ + n) * 3;
    const float px = pr[0], py = pr[1], pz = pr[2];
    const float pp = px * px + py * py + pz * pz;

    float b0 = 3.4e38f, b1 = 3.4e38f, b2 = 3.4e38f;
    int   i0 = 0, i1 = 0, i2 = 0;
    for (int m = 0; m < M_; ++m) {
        float d = pp + qq[m] - 2.0f * (px * qx[m] + py * qy[m] + pz * qz[m]);
        bool c0 = d < b0, c1 = d < b1, c2 = d < b2;
        b2 = c1 ? b1 : (c2 ? d : b2);  i2 = c1 ? i1 : (c2 ? m : i2);
        b1 = c0 ? b0 : (c1 ? d : b1);  i1 = c0 ? i0 : (c1 ? m : i1);
        b0 = c0 ? d  : b0;             i0 = c0 ? m  : i0;
    }
    float s0 = fmaxf(b0, 1e-10f), s1 = fmaxf(b1, 1e-10f), s2 = fmaxf(b2, 1e-10f);
    float w0 = 1.0f / s0, w1 = 1.0f / s1, w2 = 1.0f / s2;
    float inv = 1.0f / (w0 + w1 + w2);
    size_t base = ((size_t)b * N_ + n) * 3;
    idxO[base + 0] = i0; idxO[base + 1] = i1; idxO[base + 2] = i2;
    wO[base + 0] = w0 * inv; wO[base + 1] = w1 * inv; wO[base + 2] = w2 * inv;
}

// ---------------------------------------------------------------------------
// Transpose x [B,C,M] -> xT [B,M,C] so neighbor gathers read contiguous rows.
// grid = (M/32, C/32, B), block = 256 (32x8)
// ---------------------------------------------------------------------------
__global__ __launch_bounds__(256) void fp_transpose_x(
    const float* __restrict__ x, float* __restrict__ xT)
{
    __shared__ float tile[32][33];
    const int b  = blockIdx.z;
    const int c0 = blockIdx.y * 32, m0 = blockIdx.x * 32;
    const int tx = threadIdx.x & 31, ty = threadIdx.x >> 5;
    #pragma unroll
    for (int i = 0; i < 32; i += 8)
        tile[ty + i][tx] = x[((size_t)b * C_ + c0 + ty + i) * M_ + m0 + tx];
    __syncthreads();
    #pragma unroll
    for (int i = 0; i < 32; i += 8)
        xT[((size_t)b * M_ + m0 + ty + i) * C_ + c0 + tx] = tile[tx][ty + i];
}

// ---------------------------------------------------------------------------
// Weighted interpolation: h0[c, b*N+n] = sum_k w_k * xT[b, idx_k, c].
// Block covers 32 positions of one batch; each wave handles 4 positions and
// streams three contiguous 1 KB feature rows. Output staged in LDS and
// written row-major (c-major) to the [C, P] activation buffer.
// grid = (N/32, B), block = 256
// ---------------------------------------------------------------------------
__global__ __launch_bounds__(256) void fp_interp(
    const float* __restrict__ xT, const int* __restrict__ idx,
    const float* __restrict__ wgt, float* __restrict__ h0)
{
    __shared__ float tile[C_][33];
    const int b = blockIdx.y;
    const int n0 = blockIdx.x * 32;
    const int wave = threadIdx.x >> 5, lane = threadIdx.x & 31;

    #pragma unroll
    for (int pi = 0; pi < 4; ++pi) {
        const int pos = wave * 4 + pi;
        const size_t pb = ((size_t)b * N_ + n0 + pos) * 3;
        const int   j0 = idx[pb + 0], j1 = idx[pb + 1], j2 = idx[pb + 2];
        const float w0 = wgt[pb + 0], w1 = wgt[pb + 1], w2 = wgt[pb + 2];
        const float* r0 = xT + ((size_t)b * M_ + j0) * C_ + lane * 8;
        const float* r1 = xT + ((size_t)b * M_ + j1) * C_ + lane * 8;
        const float* r2 = xT + ((size_t)b * M_ + j2) * C_ + lane * 8;
        float4 a0 = ((const float4*)r0)[0], a1 = ((const float4*)r0)[1];
        float4 b0v = ((const float4*)r1)[0], b1v = ((const float4*)r1)[1];
        float4 c0v = ((const float4*)r2)[0], c1v = ((const float4*)r2)[1];
        float v[8];
        v[0] = w0*a0.x + w1*b0v.x + w2*c0v.x;  v[1] = w0*a0.y + w1*b0v.y + w2*c0v.y;
        v[2] = w0*a0.z + w1*b0v.z + w2*c0v.z;  v[3] = w0*a0.w + w1*b0v.w + w2*c0v.w;
        v[4] = w0*a1.x + w1*b1v.x + w2*c1v.x;  v[5] = w0*a1.y + w1*b1v.y + w2*c1v.y;
        v[6] = w0*a1.z + w1*b1v.z + w2*c1v.z;  v[7] = w0*a1.w + w1*b1v.w + w2*c1v.w;
        #pragma unroll
        for (int j = 0; j < 8; ++j) tile[lane * 8 + j][pos] = v[j];
    }
    __syncthreads();
    const int c = threadIdx.x;                   // 0..255
    float* dst = h0 + (size_t)c * P_ + (size_t)b * N_ + n0;
    #pragma unroll
    for (int j = 0; j < 32; ++j) dst[j] = tile[c][j];
}

// ---------------------------------------------------------------------------
// fp32 WMMA GEMM layer: Z[o,p] = sum_c W[o,c] * act(Hin[c,p]),
// act = relu(h*scale[c]+shift[c]) when inScale != null (fused BN of the
// previous layer), identity otherwise. Also accumulates per-channel
// sum / sum-of-squares for this layer's training-mode BatchNorm.
// Bias is skipped: it cancels exactly under (h - mean).
//
// Block tile: 64 (o) x 256 (n); 8 waves in a 2x4 grid, each wave a 32x64
// tile via 2x4 V_WMMA_F32_16X16X4_F32 accumulators. K staged in 32-chunks.
// H tile is stored K-pair-interleaved (Hlds[k/2][n][2]) so each B fragment
// is a single aligned ds_load_b64; k2-stride padded to 32 mod 64 banks so
// the two lane halves (K and K+1) read disjoint bank windows.
// grid = (P/256, O/64), block = 256
// ---------------------------------------------------------------------------
__global__ __launch_bounds__(256) void fp_gemm_bn(
    const float* __restrict__ Hin, const float* __restrict__ Wm,
    const float* __restrict__ inScale, const float* __restrict__ inShift,
    float* __restrict__ Zout, float* __restrict__ statSum,
    float* __restrict__ statSq, int Cin)
{
    __shared__ float Wlds[64][34];      // stride 34 -> conflict-free A-frag reads
    __shared__ float Hlds[16][272][2];  // [k/2][n(+16 pad)][K-pair]
    __shared__ float sSum[64];
    __shared__ float sSq[64];

    const int t = threadIdx.x;
    const int lane = t & 31, wave = t >> 5;
    const int waveO = wave >> 2, waveN = wave & 3;
    const int laneLo = lane & 15, laneHi = lane >> 4;
    const int oBlock = blockIdx.y * 64;
    const int nBlock = blockIdx.x * 256;
    const bool useBn = (inScale != nullptr);

    if (t < 64) { sSum[t] = 0.0f; sSq[t] = 0.0f; }

    v8f acc[2][4] = {};

    for (int kc = 0; kc < Cin; kc += 32) {
        __syncthreads();
        { // stage W tile: 64 rows x 32 k-cols (8 floats / thread)
            const int row = t >> 2, col = (t & 3) * 8;
            const float* src = Wm + (size_t)(oBlock + row) * Cin + kc + col;
            float4 v0 = ((const float4*)src)[0];
            float4 v1 = ((const float4*)src)[1];
            Wlds[row][col + 0] = v0.x; Wlds[row][col + 1] = v0.y;
            Wlds[row][col + 2] = v0.z; Wlds[row][col + 3] = v0.w;
            Wlds[row][col + 4] = v1.x; Wlds[row][col + 5] = v1.y;
            Wlds[row][col + 6] = v1.z; Wlds[row][col + 7] = v1.w;
        }
        { // stage H tile 32 x 256, K-pair interleaved, fusing prev BN+ReLU.
          // k2 is wave-uniform; adjacent lanes -> adjacent n (b128 stores
          // cover consecutive LDS words: conflict-free).
            #pragma unroll
            for (int i = 0; i < 8; ++i) {
                const int k2 = i * 2 + (t >> 7);
                const int n  = (t & 127) * 2;
                const int cg0 = kc + 2 * k2, cg1 = cg0 + 1;
                const float2 u0 = *(const float2*)(Hin + (size_t)cg0 * P_ + nBlock + n);
                const float2 u1 = *(const float2*)(Hin + (size_t)cg1 * P_ + nBlock + n);
                float e0 = u0.x, e1 = u0.y, f0 = u1.x, f1 = u1.y;
                if (useBn) {
                    const float s0 = inScale[cg0], h0 = inShift[cg0];
                    const float s1 = inScale[cg1], h1 = inShift[cg1];
                    e0 = fmaxf(fmaf(e0, s0, h0), 0.0f);
                    e1 = fmaxf(fmaf(e1, s0, h0), 0.0f);
                    f0 = fmaxf(fmaf(f0, s1, h1), 0.0f);
                    f1 = fmaxf(fmaf(f1, s1, h1), 0.0f);
                }
                v4f pk = {e0, f0, e1, f1};   // (H[2k2][n],H[2k2+1][n],H[2k2][n+1],H[2k2+1][n+1])
                *(v4f*)&Hlds[k2][n][0] = pk;
            }
        }
        __syncthreads();

        #pragma unroll
        for (int ks = 0; ks < 8; ++ks) {
            // A-fragment (W, 16x4 f32): lanes 0-15 K={ks*4+0,1},
            // lanes 16-31 K={ks*4+2,3}, M = lane&15. Single b64 each.
            const int acol = ks * 4 + laneHi * 2;
            const v2f a0 = *(const v2f*)&Wlds[waveO * 32 + laneLo][acol];
            const v2f a1 = *(const v2f*)&Wlds[waveO * 32 + 16 + laneLo][acol];
            // B-fragment (H, 4x16 f32): low lanes K={0,1}, high lanes K={2,3},
            // N = lane&15. Pair-interleaved -> single b64 each.
            const int k2 = ks * 2 + laneHi;
            const int nw = waveN * 64 + laneLo;
            const v2f b0 = *(const v2f*)&Hlds[k2][nw][0];
            const v2f b1 = *(const v2f*)&Hlds[k2][nw + 16][0];
            const v2f b2 = *(const v2f*)&Hlds[k2][nw + 32][0];
            const v2f b3 = *(const v2f*)&Hlds[k2][nw + 48][0];

            acc[0][0] = __builtin_amdgcn_wmma_f32_16x16x4_f32(false, a0, false, b0, (short)0, acc[0][0], false, false);
            acc[0][1] = __builtin_amdgcn_wmma_f32_16x16x4_f32(false, a0, false, b1, (short)0, acc[0][1], false, false);
            acc[0][2] = __builtin_amdgcn_wmma_f32_16x16x4_f32(false, a0, false, b2, (short)0, acc[0][2], false, false);
            acc[0][3] = __builtin_amdgcn_wmma_f32_16x16x4_f32(false, a0, false, b3, (short)0, acc[0][3], false, false);
            acc[1][0] = __builtin_amdgcn_wmma_f32_16x16x4_f32(false, a1, false, b0, (short)0, acc[1][0], false, false);
            acc[1][1] = __builtin_amdgcn_wmma_f32_16x16x4_f32(false, a1, false, b1, (short)0, acc[1][1], false, false);
            acc[1][2] = __builtin_amdgcn_wmma_f32_16x16x4_f32(false, a1, false, b2, (short)0, acc[1][2], false, false);
            acc[1][3] = __builtin_amdgcn_wmma_f32_16x16x4_f32(false, a1, false, b3, (short)0, acc[1][3], false, false);
        }
    }

    // Store Z (C/D layout: VGPR r -> M=r (lanes 0-15) / M=8+r (lanes 16-31))
    const int oW = oBlock + waveO * 32;
    const int nW = nBlock + waveN * 64;
    #pragma unroll
    for (int so = 0; so < 2; ++so) {
        #pragma unroll
        for (int sn = 0; sn < 4; ++sn) {
            #pragma unroll
            for (int r = 0; r < 8; ++r) {
                Zout[(size_t)(oW + so * 16 + r + 8 * laneHi) * P_
                     + nW + sn * 16 + laneLo] = acc[so][sn][r];
            }
        }
    }

    // Per-channel partial sums over this block's 256 n-columns:
    // butterfly across each 16-lane half (same M), then LDS + global atomics.
    #pragma unroll
    for (int so = 0; so < 2; ++so) {
        #pragma unroll
        for (int r = 0; r < 8; ++r) {
            float s = acc[so][0][r] + acc[so][1][r] + acc[so][2][r] + acc[so][3][r];
            float q2 = acc[so][0][r] * acc[so][0][r] + acc[so][1][r] * acc[so][1][r]
                     + acc[so][2][r] * acc[so][2][r] + acc[so][3][r] * acc[so][3][r];
            #pragma unroll
            for (int msk = 1; msk < 16; msk <<= 1) {
                s  += __shfl_xor(s, msk, 32);
                q2 += __shfl_xor(q2, msk, 32);
            }
            if (laneLo == 0) {
                const int oc = waveO * 32 + so * 16 + r + 8 * laneHi;
                atomicAdd(&sSum[oc], s);
                atomicAdd(&sSq[oc], q2);
            }
        }
    }
    __syncthreads();
    if (t < 64) {
        atomicAdd(statSum + oBlock + t, sSum[t]);
        atomicAdd(statSq + oBlock + t, sSq[t]);
    }
}

// ---------------------------------------------------------------------------
// BN finalize: scale = g * rsqrt(var + eps), shift = beta - mean*scale.
// ---------------------------------------------------------------------------
__global__ void fp_finalize_bn(
    const float* __restrict__ sum, const float* __restrict__ sq,
    const float* __restrict__ g, const float* __restrict__ beta,
    float* __restrict__ sc, float* __restrict__ sh, int O)
{
    int t = threadIdx.x;
    if (t < O) {
        const float inv = 1.0f / (float)P_;
        float mean = sum[t] * inv;
        float var  = sq[t] * inv - mean * mean;
        float s = g[t] * rsqrtf(var + BN_EPS);
        sc[t] = s;
        sh[t] = beta[t] - mean * s;
    }
}

// ---------------------------------------------------------------------------
// Final BN+ReLU apply: z[O2, P] -> out[B, O2, N].
// grid = 8192, block = 256; one float4 quad per thread.
// ---------------------------------------------------------------------------
__global__ __launch_bounds__(256) void fp_apply_out(
    const float* __restrict__ z, const float* __restrict__ sc,
    const float* __restrict__ sh, float* __restrict__ out)
{
    const size_t t = (size_t)blockIdx.x * 256 + threadIdx.x;  // quad index
    const int o = (int)(t >> 14);
    const int p = ((int)t & 16383) * 4;
    const int b = p >> 13, n = p & (N_ - 1);
    float4 v = *(const float4*)(z + (size_t)o * P_ + p);
    const float s = sc[o], h = sh[o];
    v.x = fmaxf(fmaf(v.x, s, h), 0.0f);
    v.y = fmaxf(fmaf(v.y, s, h), 0.0f);
    v.z = fmaxf(fmaf(v.z, s, h), 0.0f);
    v.w = fmaxf(fmaf(v.w, s, h), 0.0f);
    *(float4*)(out + (size_t)b * (O2_ * N_) + (size_t)o * N_ + n) = v;
}

// ---------------------------------------------------------------------------
// Launch
// ---------------------------------------------------------------------------
extern "C" void kernel_launch(void* const* d_in, const int* in_sizes, int n_in,
                              void* d_out, int out_size, void* d_ws, size_t ws_size,
                              hipStream_t stream)
{
    (void)in_sizes; (void)n_in; (void)out_size; (void)ws_size;

    const float* p     = (const float*)d_in[0];
    const float* q     = (const float*)d_in[1];
    const float* x     = (const float*)d_in[2];
    const float* w0    = (const float*)d_in[3];
    const float* g0    = (const float*)d_in[5];
    const float* beta0 = (const float*)d_in[6];
    const float* w1    = (const float*)d_in[7];
    const float* g1    = (const float*)d_in[9];
    const float* beta1 = (const float*)d_in[10];
    const float* w2    = (const float*)d_in[11];
    const float* g2    = (const float*)d_in[13];
    const float* beta2 = (const float*)d_in[14];
    float* out = (float*)d_out;

    // Workspace layout (floats). bufA/bufB ping-pong (64 MB each) stay L2-
    // resident; xT/idx/wgt alias bufB since they are dead before layer 0
    // writes it.
    float* ws   = (float*)d_ws;
    float* bufA = ws;                       // 16,777,216 floats
    float* bufB = ws + 16777216;            // 16,777,216 floats
    float* xT   = bufB;                     // [B,M,C] = 2,097,152 floats
    int*   idx  = (int*)(bufB + 2097152);   // 196,608 ints
    float* wgt  = bufB + 2097152 + 196608;  // 196,608 floats
    float* stats = ws + 2 * 16777216;       // 1536 stat + 1536 scale/shift
    float *sum0 = stats,        *sq0 = stats + 256;
    float *sum1 = stats + 512,  *sq1 = stats + 768;
    float *sum2 = stats + 1024, *sq2 = stats + 1280;
    float* scsh = stats + 1536;
    float *sc0 = scsh,        *sh0 = scsh + 256;
    float *sc1 = scsh + 512,  *sh1 = scsh + 768;
    float *sc2 = scsh + 1024, *sh2 = scsh + 1280;

    fp_zero_stats<<<6, 256, 0, stream>>>(stats);
    fp_knn<<<dim3(N_ / 256, B_), 256, 0, stream>>>(p, q, idx, wgt);
    fp_copy_q<<<24, 256, 0, stream>>>(q, out);
    fp_transpose_x<<<dim3(M_ / 32, C_ / 32, B_), 256, 0, stream>>>(x, xT);
    fp_interp<<<dim3(N_ / 32, B_), 256, 0, stream>>>(xT, idx, wgt, bufA);

    // Layer 0: 256 -> 256 (no input BN)
    fp_gemm_bn<<<dim3(P_ / 256, 4), 256, 0, stream>>>(bufA, w0, nullptr, nullptr,
                                                      bufB, sum0, sq0, 256);
    fp_finalize_bn<<<1, 256, 0, stream>>>(sum0, sq0, g0, beta0, sc0, sh0, 256);
    // Layer 1: 256 -> 256 (fused BN0+ReLU on input)
    fp_gemm_bn<<<dim3(P_ / 256, 4), 256, 0, stream>>>(bufB, w1, sc0, sh0,
                                                      bufA, sum1, sq1, 256);
    fp_finalize_bn<<<1, 256, 0, stream>>>(sum1, sq1, g1, beta1, sc1, sh1, 256);
    // Layer 2: 256 -> 128 (fused BN1+ReLU on input)
    fp_gemm_bn<<<dim3(P_ / 256, 2), 256, 0, stream>>>(bufA, w2, sc1, sh1,
                                                      bufB, sum2, sq2, 256);
    fp_finalize_bn<<<1, 256, 0, stream>>>(sum2, sq2, g2, beta2, sc2, sh2, 128);
    // BN2 + ReLU -> d_out (after the q copy region)
    fp_apply_out<<<8192, 256, 0, stream>>>(bufB, sc2, sh2, out + B_ * M_ * 3);
}